// MSDeformCrossAttn_68075231642067
// MI455X (gfx1250) — compile-verified
//
#include <hip/hip_runtime.h>
#include <math.h>

#define N_HEADS  8
#define N_LEVELS 4
#define N_POINTS 4
#define EMBED    256
#define HEAD_DIM 32

typedef float v2f __attribute__((ext_vector_type(2)));
typedef float v8f __attribute__((ext_vector_type(8)));

// ---------------------------------------------------------------------------
// C[M,N] = A[M,K] @ W[K,N] + bias[N], fp32 via V_WMMA_F32_16X16X4_F32.
//
// Block = 256 threads = 8 waves stacked along M; each wave owns a 16(M)x64(N)
// strip (4 accumulators, A fragment reused x4). Grid = (N/64, M/128).
// Requires M%128==0, N%64==0, K%4==0 (true for all calls) -> EXEC all-1s.
//
// W strip for the block (K=256 x N=64 = 64 KB) is staged once into LDS in
// fragment-ready order [ks][t][lane][j], so each B fragment is one
// ds_load_b64 at an immediate offset (<= 65528, fits the 16-bit DS offset),
// bank-conflict-free (lane L -> banks 2L,2L+1).
//
// Fragment layouts (CDNA5 ISA 7.12.2, wave32):
//   A 16x4:  lane L holds A[rowBase + L%16][4ks + 2*(L/16) + j], j=0..1
//   B 4x16:  lane L holds W[4ks + 2*(L/16) + j][colBase + 16t + L%16]
//   C 16x16: lane L, vgpr i holds C[rowBase + 8*(L/16) + i][colBase + L%16]
// ---------------------------------------------------------------------------
__global__ __launch_bounds__(256)
void gemm_f32_wmma(const float* __restrict__ A, const float* __restrict__ W,
                   const float* __restrict__ bias, float* __restrict__ C,
                   int M, int N, int K)
{
    __shared__ float wlds[256 / 4 * 4 * 32 * 2];      // 16384 floats = 64 KB

    const int tid  = threadIdx.x;
    const int lane = tid & 31;
    const int wave = tid >> 5;
    const int lmod = lane & 15;
    const int lhi  = lane >> 4;                       // 0 or 1
    const int row_base = (blockIdx.y * 8 + wave) * 16;
    const int col_base = blockIdx.x * 64;

    // ---- cooperative fill: global (coalesced rows) -> LDS (fragment order)
    for (int i = tid; i < K * 64; i += 256) {
        const int k = i >> 6;                         // 0..K-1
        const int n = i & 63;                         // 0..63 within strip
        const float val = W[(size_t)k * N + col_base + n];
        const int ks  = k >> 2, kr = k & 3;
        const int jlo = kr & 1, jhi = kr >> 1;        // j, lhi of this element
        const int t   = n >> 4, nm = n & 15;
        const int dlane = (jhi << 4) | nm;
        wlds[(((ks * 4 + t) * 32 + dlane) << 1) | jlo] = val;
    }
    __syncthreads();

    v8f acc0 = (v8f)(0.0f), acc1 = (v8f)(0.0f), acc2 = (v8f)(0.0f), acc3 = (v8f)(0.0f);

    const float* arow = A + (size_t)(row_base + lmod) * K + 2 * lhi;
    const v2f*   bl   = (const v2f*)wlds;             // index: (ks*4 + t)*32 + lane

    for (int ks = 0; ks < K / 4; ++ks) {
        const v2f a  = *(const v2f*)(arow + ks * 4);  // 8B aligned
        const v2f b0 = bl[(ks * 4 + 0) * 32 + lane];
        const v2f b1 = bl[(ks * 4 + 1) * 32 + lane];
        const v2f b2 = bl[(ks * 4 + 2) * 32 + lane];
        const v2f b3 = bl[(ks * 4 + 3) * 32 + lane];

        acc0 = __builtin_amdgcn_wmma_f32_16x16x4_f32(false, a, false, b0, (short)0, acc0, false, false);
        acc1 = __builtin_amdgcn_wmma_f32_16x16x4_f32(false, a, false, b1, (short)0, acc1, false, false);
        acc2 = __builtin_amdgcn_wmma_f32_16x16x4_f32(false, a, false, b2, (short)0, acc2, false, false);
        acc3 = __builtin_amdgcn_wmma_f32_16x16x4_f32(false, a, false, b3, (short)0, acc3, false, false);
    }

    v8f accs[4] = {acc0, acc1, acc2, acc3};
    #pragma unroll
    for (int t = 0; t < 4; ++t) {
        const int col = col_base + t * 16 + lmod;
        const float bv = bias[col];
        #pragma unroll
        for (int i = 0; i < 8; ++i) {
            const int row = row_base + 8 * lhi + i;
            C[(size_t)row * N + col] = accs[t][i] + bv;
        }
    }
}

// ---------------------------------------------------------------------------
// One wave32 per (b, q, head); lane = head-dim channel (0..31).
// Softmax over 16 logits + all location math is wave-uniform (redundant VALU).
// v is laid out (B, Nv, heads, 32) so each bilinear corner is one coalesced
// 128-byte wave read; the whole 22 MB value tensor is L2-resident (192 MB L2).
// ---------------------------------------------------------------------------
__global__ __launch_bounds__(256)
void msdeform_sample(const float* __restrict__ v,     // (B*Nv, 8, 32)
                     const float* __restrict__ off,   // (B*Nq, 8, 4, 4, 2)
                     const float* __restrict__ aw,    // (B*Nq, 8, 16) logits
                     const float* __restrict__ rp,    // (B*Nq, 4, 2)
                     float* __restrict__ out,         // (B*Nq, 8, 32)
                     int B, int Nq, int Nv)
{
    const int lane = threadIdx.x & 31;
    const int wid  = (blockIdx.x * blockDim.x + threadIdx.x) >> 5;
    const int h  = wid % N_HEADS;
    const int bq = wid / N_HEADS;                     // = b*Nq + q
    const int b  = bq / Nq;

    // softmax over the 16 (level, point) logits
    const float* awp = aw + ((size_t)bq * N_HEADS + h) * (N_LEVELS * N_POINTS);
    float wgt[16];
    float mx = -3.0e38f;
    #pragma unroll
    for (int i = 0; i < 16; ++i) { wgt[i] = awp[i]; mx = fmaxf(mx, wgt[i]); }
    float s = 0.0f;
    #pragma unroll
    for (int i = 0; i < 16; ++i) { wgt[i] = __expf(wgt[i] - mx); s += wgt[i]; }
    const float inv = 1.0f / s;

    const float* offp = off + ((size_t)bq * N_HEADS + h) * (N_LEVELS * N_POINTS * 2);
    const float* rpp  = rp + (size_t)bq * (N_LEVELS * 2);

    const int Hs[4]  = {64, 32, 16, 8};
    const int Ws[4]  = {64, 32, 16, 8};
    const int cur[4] = {0, 4096, 5120, 5376};

    float acc = 0.0f;
    #pragma unroll
    for (int l = 0; l < N_LEVELS; ++l) {
        const int   Hl = Hs[l], Wl = Ws[l];
        const float rx = rpp[l * 2 + 0], ry = rpp[l * 2 + 1];
        const float* vbase =
            v + ((size_t)(b * Nv + cur[l]) * N_HEADS + h) * HEAD_DIM + lane;

        #pragma unroll
        for (int p = 0; p < N_POINTS; ++p) {
            const float ox = offp[(l * N_POINTS + p) * 2 + 0];
            const float oy = offp[(l * N_POINTS + p) * 2 + 1];
            // loc in [0,1]; align_corners=False pixel coords: x*W - 0.5
            const float px = (rx + ox / (float)Wl) * (float)Wl - 0.5f;
            const float py = (ry + oy / (float)Hl) * (float)Hl - 0.5f;
            const float x0f = floorf(px), y0f = floorf(py);
            const float wx = px - x0f, wy = py - y0f;
            const int   x0 = (int)x0f,  y0 = (int)y0f;
            const float w  = wgt[l * N_POINTS + p] * inv;

            float sample = 0.0f;
            #pragma unroll
            for (int c = 0; c < 4; ++c) {
                const int dx = c & 1, dy = c >> 1;
                const int xi = x0 + dx, yi = y0 + dy;
                if (xi >= 0 && xi < Wl && yi >= 0 && yi < Hl) {  // wave-uniform
                    const float cw = (dx ? wx : 1.0f - wx) * (dy ? wy : 1.0f - wy);
                    sample += cw * vbase[(size_t)(yi * Wl + xi) * (N_HEADS * HEAD_DIM)];
                }
            }
            acc += w * sample;
        }
    }
    out[((size_t)bq * N_HEADS + h) * HEAD_DIM + lane] = acc;
}

// ---------------------------------------------------------------------------
// inputs: 0 query (B,Nq,256)        1 value (B,Nv,256)    2 reference_points
//         3 value_proj_w  4 value_proj_b  5 offsets_w  6 offsets_b
//         7 attn_w_w      8 attn_w_b      9 out_w     10 out_b
// ---------------------------------------------------------------------------
extern "C" void kernel_launch(void* const* d_in, const int* in_sizes, int n_in,
                              void* d_out, int out_size, void* d_ws, size_t ws_size,
                              hipStream_t stream) {
    const float* query   = (const float*)d_in[0];
    const float* value   = (const float*)d_in[1];
    const float* refpts  = (const float*)d_in[2];
    const float* vproj_w = (const float*)d_in[3];
    const float* vproj_b = (const float*)d_in[4];
    const float* off_w   = (const float*)d_in[5];
    const float* off_b   = (const float*)d_in[6];
    const float* aw_w    = (const float*)d_in[7];
    const float* aw_b    = (const float*)d_in[8];
    const float* out_w   = (const float*)d_in[9];
    const float* out_b   = (const float*)d_in[10];
    float* out = (float*)d_out;

    const int B  = 4;
    const int Nq = 5440;                 // 64^2 + 32^2 + 16^2 + 8^2
    const int Nv = 5440;
    const int M  = B * Nq;               // 21760, divisible by 128

    // workspace: v(M*256) | off(M*256) | aw(M*128) | attn(M*256)  ~78 MB
    char*  ws       = (char*)d_ws;
    float* v_buf    = (float*)(ws);
    float* off_buf  = (float*)(ws + (size_t)M * 256 * sizeof(float));
    float* aw_buf   = (float*)(ws + (size_t)M * 512 * sizeof(float));
    float* attn_buf = (float*)(ws + (size_t)M * 640 * sizeof(float));

    const dim3 blk(256);
    const dim3 g256(EMBED / 64, M / 128);   // N=256
    const dim3 g128(128 / 64,  M / 128);    // N=128

    // 1) v = value @ Wv + bv            (M,256)x(256,256)
    gemm_f32_wmma<<<g256, blk, 0, stream>>>(value, vproj_w, vproj_b, v_buf, M, 256, 256);
    // 2) off = query @ Woff + boff      (M,256)x(256,256)
    gemm_f32_wmma<<<g256, blk, 0, stream>>>(query, off_w, off_b, off_buf, M, 256, 256);
    // 3) aw logits = query @ Waw + baw  (M,256)x(256,128)
    gemm_f32_wmma<<<g128, blk, 0, stream>>>(query, aw_w, aw_b, aw_buf, M, 128, 256);

    // 4) softmax + deformable bilinear sampling + weighted sum
    const int nwaves = M * N_HEADS;         // 174080, divisible by 8
    msdeform_sample<<<dim3(nwaves / 8), blk, 0, stream>>>(
        v_buf, off_buf, aw_buf, refpts, attn_buf, B, Nq, Nv);

    // 5) out = attn @ Wout + bout       (M,256)x(256,256)
    gemm_f32_wmma<<<g256, blk, 0, stream>>>(attn_buf, out_w, out_b, out, M, 256, 256);
}